// RNN_72541997629806
// MI455X (gfx1250) — compile-verified
//
#include <hip/hip_runtime.h>
#include <hip/hip_bf16.h>

// ---------------------------------------------------------------------------
// 5-layer stacked LSTM (H=4, SEQ=610, B=128) + Linear(4->1) + FC(610->5)
//
// MI455X (gfx1250, wave32) design, v4:
//  * gates^T = W[16x4] x in[4x16] on V_WMMA_F32_16X16X4_F32 (fp32; problem is
//    latency-bound at ~0.1 GFLOP total, matrix-core shape match + precision).
//  * Gate rows permuted [i0,f0,g0,o0,i1,..,o1 | i2,..,o3]: each lane gets a
//    complete (i,f,g,o) set for 2 hidden units of its own batch column, and
//    the new (hA,hB) pair is already in next-WMMA B-operand layout ->
//    ZERO cross-lane ops on the recurrence critical path.
//  * Per layer: hidden-GEMM FIRST (depends only on prev-timestep h -> issues
//    during previous layer's activation chain), input-GEMM second (the only
//    op waiting on the serial inter-layer dependency).
//  * Native v_tanh_f32 for tanh AND sigmoid (sig(x)=0.5*tanh(0.5x)+0.5).
//  * t-loop unrolled x2 (610=2*305): halves exposed ds_load latency (the
//    scheduler rotates prefetches to the head; with unroll the 2nd sub-body's
//    loads are fully covered) and halves loop overhead.
//  * 8 blocks x 1 wave32 (one per WGP); x tile + w_fc staged in LDS;
//    weights/bias/state VGPR-resident; 5 cross-lane ops total (epilogue).
// ---------------------------------------------------------------------------

typedef __attribute__((ext_vector_type(2))) float v2f;
typedef __attribute__((ext_vector_type(4))) float v4f;
typedef __attribute__((ext_vector_type(8))) float v8f;

#define SEQ_T 610
#define NLAYERS 5

__device__ __forceinline__ float tanh_fast(float x) {
#if __has_builtin(__builtin_amdgcn_tanhf)
  return __builtin_amdgcn_tanhf(x);   // v_tanh_f32 (gfx1250 trans op)
#else
  float e = __expf(-2.0f * x);
  return (1.0f - e) / (1.0f + e);
#endif
}

// sigmoid(x) = (1 + tanh(x/2)) / 2 : mul -> tanh -> fma, single trans op.
__device__ __forceinline__ float sigm(float x) {
  return fmaf(0.5f, tanh_fast(0.5f * x), 0.5f);
}

// D = A(16x4) * B(4x16) + C, fp32 WMMA.
__device__ __forceinline__ v8f wmma4(v2f a, v2f b, v8f c) {
  return __builtin_amdgcn_wmma_f32_16x16x4_f32(
      /*neg_a=*/false, a, /*neg_b=*/false, b,
      /*c_mod=*/(short)0, c, /*reuse_a=*/false, /*reuse_b=*/false);
}

// Position M in the A matrix holds original PyTorch gate row
// perm(M) = (M&3)*4 + (M>>2)   (gate-major -> unit-major ordering).
__device__ __forceinline__ int perm_row(int M) { return (M & 3) * 4 + (M >> 2); }

__global__ void __launch_bounds__(32, 1)
lstm_stack_kernel(const float* __restrict__ x,       // [128,610,2]
                  const float* __restrict__ w_ih0,   // [16,2]
                  const float* __restrict__ w_ih,    // [4,16,4]
                  const float* __restrict__ w_hh,    // [5,16,4]
                  const float* __restrict__ b_ih,    // [5,16]
                  const float* __restrict__ b_hh,    // [5,16]
                  const float* __restrict__ w_lin,   // [1,4]
                  const float* __restrict__ b_lin,   // [1]
                  const float* __restrict__ w_fc,    // [5,610]
                  const float* __restrict__ b_fc,    // [5]
                  float* __restrict__ out) {         // [128,5]
  extern __shared__ float smem[];
  float* xs    = smem;                    // [16*1220] floats (contiguous tile)
  float* wfc_s = smem + 16 * (SEQ_T * 2); // [5*610]

  const int lane = threadIdx.x;           // 0..31 (wave32)
  const bool lo  = (lane & 16) == 0;      // lower lane half
  const int row  = lane & 15;             // batch column / A-row index
  const int bb   = blockIdx.x * 16;       // batch tile base

  // ---- stage x tile (rows bb..bb+15 contiguous) + w_fc into LDS -----------
  {
    const v4f* src = (const v4f*)(x + (size_t)bb * (SEQ_T * 2));
    v4f* dst = (v4f*)xs;
    for (int i = lane; i < 16 * (SEQ_T * 2) / 4; i += 32) dst[i] = src[i];
    for (int i = lane; i < 5 * SEQ_T; i += 32) wfc_s[i] = w_fc[i];
  }
  __syncthreads();

  // ---- weights in WMMA A layout (VGPR-resident), gate rows permuted -------
  // A 16x4 f32: lanes 0-15 hold K=0,1; lanes 16-31 hold K=2,3; M = lane&15.
  const int kb = lo ? 0 : 2;
  v2f Awi[NLAYERS], Awh[NLAYERS];
  v8f biasC[NLAYERS];
  const int wrow = perm_row(row);
#pragma unroll
  for (int l = 0; l < NLAYERS; ++l) {
    const float* wh = w_hh + l * 16 * 4;
    Awh[l].x = wh[wrow * 4 + kb];
    Awh[l].y = wh[wrow * 4 + kb + 1];
    if (l == 0) {
      // input size 2: K=2,3 are zero padding (upper-lane A columns = 0, so
      // the B operand's upper lanes may carry garbage -- contributes 0).
      Awi[l].x = lo ? w_ih0[wrow * 2 + 0] : 0.0f;
      Awi[l].y = lo ? w_ih0[wrow * 2 + 1] : 0.0f;
    } else {
      const float* wi = w_ih + (l - 1) * 16 * 4;
      Awi[l].x = wi[wrow * 4 + kb];
      Awi[l].y = wi[wrow * 4 + kb + 1];
    }
    // C bias: VGPR j holds permuted row j (lo) / j+8 (hi), all columns.
#pragma unroll
    for (int j = 0; j < 8; ++j) {
      int pr = perm_row(j + (lo ? 0 : 8));
      biasC[l][j] = b_ih[l * 16 + pr] + b_hh[l * 16 + pr];
    }
  }

  // Linear(4->1) weights split by lane half: lower = units 0,1; upper = 2,3.
  const float wlA = lo ? w_lin[0] : w_lin[2];
  const float wlB = lo ? w_lin[1] : w_lin[3];
  const float blh = lo ? b_lin[0] : 0.0f;   // bias folded into lower partial

  // ---- recurrent state: lane n = units 0,1 of batch n; lane n+16 = 2,3.
  float hA[NLAYERS], hB[NLAYERS], cA[NLAYERS], cB[NLAYERS];
#pragma unroll
  for (int l = 0; l < NLAYERS; ++l) { hA[l] = hB[l] = cA[l] = cB[l] = 0.0f; }

  float acc[5] = {0.0f, 0.0f, 0.0f, 0.0f, 0.0f};  // per-half FC partials

  const int xbase = row * (SEQ_T * 2);

#pragma unroll 2
  for (int t = 0; t < SEQ_T; ++t) {
    // layer-0 B operand: upper lanes carry garbage * zero A-columns.
    v2f bIn;
    bIn.x = xs[xbase + 2 * t];
    bIn.y = xs[xbase + 2 * t + 1];

#pragma unroll
    for (int l = 0; l < NLAYERS; ++l) {
      // Hidden-GEMM first: operands ready since last timestep -> overlaps
      // with the previous layer's activation chain. Input-GEMM second is
      // the only op on the serial inter-layer dependency.
      v2f bH; bH.x = hA[l]; bH.y = hB[l];   // already {h0|h2},{h1|h3} layout
      v8f d = wmma4(Awh[l], bH, biasC[l]);
      d = wmma4(Awi[l], bIn, d);

      // Lane-local gates (permuted rows): d[0..3] = i,f,g,o of unit A;
      //                                   d[4..7] = i,f,g,o of unit B.
      float ca = sigm(d[1]) * cA[l] + sigm(d[0]) * tanh_fast(d[2]);
      float cb = sigm(d[5]) * cB[l] + sigm(d[4]) * tanh_fast(d[6]);
      cA[l] = ca; cB[l] = cb;
      hA[l] = sigm(d[3]) * tanh_fast(ca);
      hB[l] = sigm(d[7]) * tanh_fast(cb);

      // next layer's input operand = this layer's new h (no lane movement)
      bIn.x = hA[l]; bIn.y = hB[l];
    }

    // Linear(4->1) per-half partial folded into FC(610->5) accumulators.
    float part = hA[NLAYERS - 1] * wlA + hB[NLAYERS - 1] * wlB + blh;
#pragma unroll
    for (int j = 0; j < 5; ++j) acc[j] += part * wfc_s[j * SEQ_T + t];
  }

  // Combine lane halves (5 cross-lane ops TOTAL) and write [16,5] outputs.
#pragma unroll
  for (int j = 0; j < 5; ++j) acc[j] += __shfl_xor(acc[j], 16, 32);
  if (lo) {
#pragma unroll
    for (int j = 0; j < 5; ++j) out[(bb + row) * 5 + j] = acc[j] + b_fc[j];
  }
}

extern "C" void kernel_launch(void* const* d_in, const int* in_sizes, int n_in,
                              void* d_out, int out_size, void* d_ws, size_t ws_size,
                              hipStream_t stream) {
  (void)in_sizes; (void)n_in; (void)d_ws; (void)ws_size; (void)out_size;
  const float* x     = (const float*)d_in[0];
  const float* w_ih0 = (const float*)d_in[1];
  const float* w_ih  = (const float*)d_in[2];
  const float* w_hh  = (const float*)d_in[3];
  const float* b_ih  = (const float*)d_in[4];
  const float* b_hh  = (const float*)d_in[5];
  const float* w_lin = (const float*)d_in[6];
  const float* b_lin = (const float*)d_in[7];
  const float* w_fc  = (const float*)d_in[8];
  const float* b_fc  = (const float*)d_in[9];
  float* out = (float*)d_out;

  const size_t lds_bytes =
      (size_t)(16 * SEQ_T * 2 + 5 * SEQ_T) * sizeof(float);
  // 128 batch rows / 16 per wave = 8 independent single-wave workgroups.
  lstm_stack_kernel<<<8, 32, lds_bytes, stream>>>(
      x, w_ih0, w_ih, w_hh, b_ih, b_hh, w_lin, b_lin, w_fc, b_fc, out);
}